// CharacterLanguageModel_31233002176717
// MI455X (gfx1250) — compile-verified
//
#include <hip/hip_runtime.h>
#include <stdint.h>

// Problem constants (from reference): B=1024, S=256, W=16, 256 classes, EMB=50
#define NUM_W     16
#define NUM_C     256
#define NUM_E     50
#define NBAGS     (1024 * 256)        // 262144 bags
#define NTILES    (NBAGS / 16)        // 16384 16-bag tiles
#define WAVES     8
#define THREADS   (WAVES * 32)
#define BLOCKS    1024

typedef __attribute__((ext_vector_type(2))) float        v2f;
typedef __attribute__((ext_vector_type(8))) float        v8f;
typedef __attribute__((ext_vector_type(4))) unsigned int u32x4;
typedef __attribute__((ext_vector_type(8))) int          i32x8;
typedef __attribute__((ext_vector_type(4))) int          i32x4;

__global__ __launch_bounds__(THREADS)
void embbag_wmma_kernel(const int* __restrict__ chars,
                        const int* __restrict__ lengths,
                        const float* __restrict__ table,
                        float* __restrict__ out)
{
    // LDS: raw f32 table (TDM target), per-wave f16 count matrices, 1/len
    __shared__ float     tab[NUM_C * NUM_E];            // 51200 B
    __shared__ _Float16  counts[WAVES][16][NUM_C];      // 65536 B
    __shared__ float     inv[WAVES][16];                //   512 B

    const int tid  = threadIdx.x;
    const int wv   = tid >> 5;
    const int lane = tid & 31;

    // ---- Stage the embedding table into LDS with the Tensor Data Mover ----
#if __has_builtin(__builtin_amdgcn_tensor_load_to_lds)
    if (tid == 0) {
        uint32_t lds_addr = (uint32_t)(uintptr_t)(&tab[0]);
        uint64_t ga       = (uint64_t)(uintptr_t)table;
        // D# group0: count=1 | lds_addr | global_addr[56:0] | type=2
        u32x4 g0 = { 1u,
                     lds_addr,
                     (uint32_t)ga,
                     (uint32_t)((ga >> 32) & 0x01FFFFFFu) | (2u << 30) };
        // D# group1: data_size=4B; tensor_dim0=50; tensor_dim1=256;
        //            tile_dim0=50; tile_dim1=256; tensor_dim0_stride=50
        i32x8 g1 = { (int)(2u << 16),        // data_size=2 (4 bytes)
                     (int)(50u << 16),       // tensor_dim0[15:0] @ bits 63:48
                     (int)(256u << 16),      // tensor_dim1[15:0] @ bits 95:80
                     (int)(50u << 16),       // tile_dim0 @ bits 127:112
                     256,                    // tile_dim1 @ bits 143:128
                     50,                     // tensor_dim0_stride low 32
                     0, 0 };
        i32x4 gz4 = { 0, 0, 0, 0 };
        i32x8 gz8 = { 0, 0, 0, 0, 0, 0, 0, 0 };
        __builtin_amdgcn_tensor_load_to_lds(g0, g1, gz4, gz4, gz8, 0);
        __builtin_amdgcn_s_wait_tensorcnt(0);
    }
#else
    for (int i = tid; i < NUM_C * NUM_E; i += THREADS) tab[i] = table[i];
#endif
    __syncthreads();

    const int m  = lane & 15;   // A-matrix row (bag in tile) / B,C,D column-in-16
    const int hi = lane >> 4;   // lane half

    for (int tile = blockIdx.x * WAVES + wv; tile < NTILES; tile += gridDim.x * WAVES) {
        const int bag0 = tile * 16;

        // ---- zero this wave's 16x256 f16 count matrix (2048 dwords) ----
        uint32_t* cz = (uint32_t*)&counts[wv][0][0];
        #pragma unroll
        for (int j = 0; j < 64; ++j) cz[j * 32 + lane] = 0u;
        __builtin_amdgcn_wave_barrier();

        // ---- build counts: lane i owns bag i (race-free serial RMW) ----
        if (lane < 16) {
            const int bag = bag0 + lane;
            const int len = lengths[bag];
            inv[wv][lane] = 1.0f / (float)len;
            const int* cp = chars + bag * NUM_W;
            for (int w = 0; w < NUM_W; ++w) {
                if (w < len) {
                    const int c = cp[w];
                    counts[wv][lane][c] += (_Float16)1.0f;
                }
            }
        }
        __builtin_amdgcn_wave_barrier();

        // ---- D[16x64] = Counts[16x256] @ Table[256x50(pad 64)] via f32 WMMA ----
        v8f acc0 = {}, acc1 = {}, acc2 = {}, acc3 = {};
        const int c0 = m;                 // N-tile 0 column
        const int c1 = m + 16;
        const int c2 = m + 32;
        const int c3 = (m + 48 < NUM_E) ? (m + 48) : (NUM_E - 1);  // clamp pad cols
        for (int kk = 0; kk < 64; ++kk) {
            const int k0 = kk * 4 + 2 * hi;   // ISA 32-bit A 16x4 layout
            v2f a;
            a.x = (float)counts[wv][m][k0];
            a.y = (float)counts[wv][m][k0 + 1];
            const float* r0 = &tab[k0 * NUM_E];
            const float* r1 = &tab[(k0 + 1) * NUM_E];
            v2f b;
            b.x = r0[c0]; b.y = r1[c0];
            acc0 = __builtin_amdgcn_wmma_f32_16x16x4_f32(false, a, false, b, (short)0, acc0, false, false);
            b.x = r0[c1]; b.y = r1[c1];
            acc1 = __builtin_amdgcn_wmma_f32_16x16x4_f32(false, a, false, b, (short)0, acc1, false, false);
            b.x = r0[c2]; b.y = r1[c2];
            acc2 = __builtin_amdgcn_wmma_f32_16x16x4_f32(false, a, false, b, (short)0, acc2, false, false);
            b.x = r0[c3]; b.y = r1[c3];
            acc3 = __builtin_amdgcn_wmma_f32_16x16x4_f32(false, a, false, b, (short)0, acc3, false, false);
        }

        // ---- scale by 1/len and store (C/D layout: VGPR v -> row v + 8*hi) ----
        #pragma unroll
        for (int v = 0; v < 8; ++v) {
            const int row = v + 8 * hi;
            const float s = inv[wv][row];
            float* op = out + (size_t)(bag0 + row) * NUM_E;
            op[m]      = acc0[v] * s;
            op[m + 16] = acc1[v] * s;
            op[m + 32] = acc2[v] * s;
            if (m + 48 < NUM_E) op[m + 48] = acc3[v] * s;
        }
    }
}

extern "C" void kernel_launch(void* const* d_in, const int* in_sizes, int n_in,
                              void* d_out, int out_size, void* d_ws, size_t ws_size,
                              hipStream_t stream) {
    const int*   chars   = (const int*)d_in[0];
    const int*   lengths = (const int*)d_in[1];
    const float* table   = (const float*)d_in[2];
    float*       out     = (float*)d_out;
    (void)in_sizes; (void)n_in; (void)out_size; (void)d_ws; (void)ws_size;
    hipLaunchKernelGGL(embbag_wmma_kernel, dim3(BLOCKS), dim3(THREADS), 0, stream,
                       chars, lengths, table, out);
}